// LinearRegressionForecast_72748156059734
// MI455X (gfx1250) — compile-verified
//
#include <hip/hip_runtime.h>

// LinearRegressionForecast: x[B=512, L=512, C=256] fp32 -> out[B, L+96, C]
// Bandwidth-bound (256 MB read + 318 MB write, ~25us floor at 23.3 TB/s).
// Reduction phase is the thin GEMM [ones; t]^T(2xL) x X_b(LxC), mapped onto
// V_WMMA_F32_16X16X4_F32 accumulating over K=L in steps of 4.

typedef __attribute__((ext_vector_type(2))) float v2f;
typedef __attribute__((ext_vector_type(4))) float f4;
typedef __attribute__((ext_vector_type(8))) float v8f;

constexpr int Bn = 512;
constexpr int Ln = 512;
constexpr int Cn = 256;
constexpr int Pn = 96;
constexpr int Tn = Ln + Pn;  // 608

__global__ __launch_bounds__(512) void linreg_forecast_kernel(
    const float* __restrict__ x, float* __restrict__ out) {
  __shared__ float s_slope[Cn];
  __shared__ float s_inter[Cn];

  const int b      = blockIdx.x;
  const int tid    = threadIdx.x;
  const int wave   = tid >> 5;   // 0..15 : 16-wide c-tile owned by this wave
  const int lane   = tid & 31;
  const int lane16 = lane & 15;
  const int half   = lane >> 4;  // lanes 16-31 hold K=2,3 per A/B VGPR layout

  // ---------------- Phase 1: WMMA reduction over L ----------------
  // A (16x4 f32): row M=0 = ones, row M=1 = time values, rows 2..15 = 0.
  // A layout: VGPR0 = K{0,2}, VGPR1 = K{1,3}; lane holds row M=lane16.
  // B (4x16 f32): B[k][n] = x[b, l+k, c0+n]; VGPR0 = rows {0,2}, VGPR1 = {1,3}.
  const float* p = x + (size_t)b * Ln * Cn + (size_t)(2 * half) * Cn
                     + wave * 16 + lane16;

  // Two interleaved accumulators (l%8==0 chunk -> acc0, l%8==4 -> acc1) to
  // break the WMMA D->C dependency chain and keep loads in flight.
  v2f a0, a1;
  a0.x = (lane16 == 0) ? 1.0f : ((lane16 == 1) ? (float)(2 * half)     : 0.0f);
  a0.y = (lane16 == 0) ? 1.0f : ((lane16 == 1) ? (float)(2 * half + 1) : 0.0f);
  a1.x = (lane16 == 0) ? 1.0f : ((lane16 == 1) ? (float)(4 + 2 * half)     : 0.0f);
  a1.y = (lane16 == 0) ? 1.0f : ((lane16 == 1) ? (float)(4 + 2 * half + 1) : 0.0f);
  const float ainc = (lane16 == 1) ? 8.0f : 0.0f;  // only the t-row advances

  v8f acc0 = {};
  v8f acc1 = {};

  #pragma unroll 2
  for (int l = 0; l < Ln; l += 8) {
    v2f bm0, bm1;
    bm0.x = p[0];           // row l + 2*half
    bm0.y = p[Cn];          // row l + 2*half + 1
    bm1.x = p[4 * Cn];      // row l + 4 + 2*half
    bm1.y = p[5 * Cn];      // row l + 4 + 2*half + 1
    acc0 = __builtin_amdgcn_wmma_f32_16x16x4_f32(false, a0, false, bm0,
                                                 (short)0, acc0, false, false);
    acc1 = __builtin_amdgcn_wmma_f32_16x16x4_f32(false, a1, false, bm1,
                                                 (short)0, acc1, false, false);
    a0.x += ainc; a0.y += ainc;
    a1.x += ainc; a1.y += ainc;
    p += 8 * Cn;
  }

  // D layout: VGPR r holds row M=r in lanes 0-15 (M=8+r in lanes 16-31).
  // Row 0 = sum_y, row 1 = sum_ty, for column n = lane16.
  if (lane < 16) {
    const float sum_y  = acc0[0] + acc1[0];
    const float sum_ty = acc0[1] + acc1[1];
    constexpr float n_inv   = 1.0f / 512.0f;
    constexpr float sum_t   = 130816.0f;           // L(L-1)/2
    constexpr float den_inv = 1.0f / 11184768.0f;  // sum_t2 - sum_t^2/n
    const float slope = (sum_ty - sum_t * sum_y * n_inv) * den_inv;
    const float inter = (sum_y - slope * sum_t) * n_inv;
    const int c = wave * 16 + lane16;
    s_slope[c] = slope;
    s_inter[c] = inter;
  }
  __syncthreads();

  // ---------------- Phase 2: stream the forecast out ----------------
  // 512 threads: 64 threads cover one 256-float row as float4s; 8 row-groups
  // walk t with stride 8. Fully coalesced 16B stores.
  const int c4   = (tid & 63) * 4;
  const int tsub = tid >> 6;  // 0..7
  const f4 sl = *(const f4*)(s_slope + c4);
  const f4 ic = *(const f4*)(s_inter + c4);
  float* ob = out + (size_t)b * Tn * Cn + c4;
  for (int t = tsub; t < Tn; t += 8) {
    const float tf = (float)t;
    f4 v;
    v.x = sl.x * tf + ic.x;
    v.y = sl.y * tf + ic.y;
    v.z = sl.z * tf + ic.z;
    v.w = sl.w * tf + ic.w;
    *(f4*)(ob + (size_t)t * Cn) = v;
  }
}

extern "C" void kernel_launch(void* const* d_in, const int* in_sizes, int n_in,
                              void* d_out, int out_size, void* d_ws, size_t ws_size,
                              hipStream_t stream) {
  const float* x = (const float*)d_in[0];
  (void)in_sizes; (void)n_in; (void)d_ws; (void)ws_size; (void)out_size;
  // d_in[1] = pred_len (==96), baked in at compile time.
  float* out = (float*)d_out;
  linreg_forecast_kernel<<<Bn, 512, 0, stream>>>(x, out);
}